// RAGQueryOptimizer_87411174408615
// MI455X (gfx1250) — compile-verified
//
#include <hip/hip_runtime.h>
#include <hip/hip_bf16.h>
#include <math.h>
#include <stdint.h>

// ---------------------------------------------------------------------------
// RAG query optimizer for MI455X (gfx1250, wave32).
// Dominant cost: 512x65536x128 fp32 GEMM for pairwise distances, done with
// V_WMMA_F32_16X16X4_F32 (fp32 matrix pipe, preserves fp32 top-k semantics).
// c_gathered (32MB) lives in the 192MB L2 across all 32 query-tile blocks.
// Ctx tiles are staged with GLOBAL_LOAD_ASYNC_TO_LDS_B128 (ASYNCcnt-tracked,
// no VGPR round-trip) into double-buffered LDS, overlapping the WMMA chain.
// ---------------------------------------------------------------------------

typedef __attribute__((ext_vector_type(2))) float v2f;
typedef __attribute__((ext_vector_type(8))) float v8f;

#define EMBD   128
#define TOPN   8
#define BATCH  4
#define LQ     128
#define LC     65536
#define NQ     (BATCH * LQ)   // 512 query rows
#define QT     16             // query rows per workgroup tile
#define CT     16             // ctx rows per WMMA tile
#define NWAVES 4
#define CSTR   132            // padded LDS stride (row step = 4 banks -> conflict-free)

// --------------------------- context gather --------------------------------
// one wave per context row: gather emb_table[token] -> c_gathered, + ||c||^2
__global__ __launch_bounds__(256) void gather_ctx(const int* __restrict__ ctok,
                                                  const float* __restrict__ emb,
                                                  float* __restrict__ cg,
                                                  float* __restrict__ cn) {
    int wave = threadIdx.x >> 5;
    int lane = threadIdx.x & 31;
    int row  = blockIdx.x * 8 + wave;
    if (row >= LC) return;
    int tok = ctok[row];
    const float4* src = (const float4*)(emb + (size_t)tok * EMBD);
    float4*       dst = (float4*)(cg + (size_t)row * EMBD);
    float4 v = src[lane];            // 32 lanes x float4 = 128 floats
    dst[lane] = v;
    float s = v.x * v.x + v.y * v.y + v.z * v.z + v.w * v.w;
    #pragma unroll
    for (int off = 16; off > 0; off >>= 1) s += __shfl_xor(s, off, 32);
    if (lane == 0) cn[row] = s;
}

// --------------------------- query gather + qp -----------------------------
// one block per query row: gather q, ||q||^2, and qp = q @ W + b
__global__ __launch_bounds__(EMBD) void gather_q(const int* __restrict__ qtok,
                                                 const float* __restrict__ emb,
                                                 const float* __restrict__ W,
                                                 const float* __restrict__ bias,
                                                 float* __restrict__ qg,
                                                 float* __restrict__ qn,
                                                 float* __restrict__ qp) {
    __shared__ float qsh[EMBD];
    __shared__ float red[EMBD];
    int i = blockIdx.x;       // 0..511
    int d = threadIdx.x;      // 0..127
    int tok = qtok[i];
    float v = emb[(size_t)tok * EMBD + d];
    qg[(size_t)i * EMBD + d] = v;
    qsh[d] = v;
    red[d] = v * v;
    __syncthreads();
    #pragma unroll
    for (int off = 64; off > 0; off >>= 1) {
        if (d < off) red[d] += red[d + off];
        __syncthreads();
    }
    if (d == 0) qn[i] = red[0];
    float acc = bias[d];
    #pragma unroll 8
    for (int k = 0; k < EMBD; ++k) acc = fmaf(qsh[k], W[k * EMBD + d], acc);
    qp[(size_t)i * EMBD + d] = acc;
}

// ---- async stage one 16x128 ctx tile into (padded) LDS: 16 x b128/lane ----
// Each issue moves one 128-float row (32 lanes x 16B), fully coalesced.
__device__ __forceinline__ void async_tile_load(const float* __restrict__ gbase,
                                                const float* lds_base, int lane) {
    #pragma unroll
    for (int r = 0; r < CT; ++r) {
        uint64_t ga = (uint64_t)(uintptr_t)(gbase + r * EMBD + lane * 4);
        uint32_t la = (uint32_t)(uintptr_t)(lds_base + r * CSTR + lane * 4);
        asm volatile("global_load_async_to_lds_b128 %0, %1, off"
                     :: "v"(la), "v"(ga) : "memory");
    }
}

// --------------------------- main distance/top-k kernel --------------------
__global__ __launch_bounds__(128) void rag_main(const float* __restrict__ cg,
                                                const float* __restrict__ cn,
                                                const float* __restrict__ qg,
                                                const float* __restrict__ qn,
                                                const float* __restrict__ qp,
                                                float* __restrict__ importance) {
    __shared__ float q_sh[QT][CSTR];                 // query tile (A operand)
    __shared__ float qp_sh[QT][EMBD];                // projected queries
    __shared__ float qn_sh[QT];
    __shared__ float c_sh[NWAVES][2][CT][CSTR];      // per-wave double-buffered ctx
    __shared__ float s_sh[NWAVES][QT][17];           // per-wave distance tile
    __shared__ float mv[QT][64];                     // merge: 4 waves * 2 lanes * 8
    __shared__ int   mi[QT][64];
    __shared__ float tvv[QT][TOPN];
    __shared__ int   tii[QT][TOPN];
    __shared__ float lg[QT][TOPN];

    const int tid   = threadIdx.x;
    const int wave  = tid >> 5;
    const int lane  = tid & 31;
    const int half  = lane >> 4;    // which 16-lane half
    const int m16   = lane & 15;
    const int qbase = blockIdx.x * QT;

    // stage query tile + projected queries + norms (cross-wave: q_sh shared)
    for (int idx = tid; idx < QT * EMBD; idx += 128) {
        int r = idx >> 7, col = idx & (EMBD - 1);
        q_sh[r][col]  = qg[(size_t)(qbase + r) * EMBD + col];
        qp_sh[r][col] = qp[(size_t)(qbase + r) * EMBD + col];
    }
    if (tid < QT) qn_sh[tid] = qn[qbase + tid];
    __syncthreads();

    // per-lane running top-8 (ascending sorted; [7] = current worst kept)
    float topv[TOPN];
    int   topi[TOPN];
    #pragma unroll
    for (int j = 0; j < TOPN; ++j) { topv[j] = 3.4e38f; topi[j] = 0; }

    const int ntiles = (LC / CT) / NWAVES;           // 1024 tiles per wave
    const int cbase0 = wave * (LC / NWAVES);

    // prologue: kick off async fill of buffer 0 with tile 0
    async_tile_load(cg + (size_t)cbase0 * EMBD, &c_sh[wave][0][0][0], lane);

    for (int t = 0; t < ntiles; ++t) {
        const int cbase = cbase0 + t * CT;
        const int cur   = t & 1;

        // issue next tile into the other buffer (its ds_load consumers from
        // iteration t-1 are all complete: their data fed WMMAs already issued)
        if (t + 1 < ntiles) {
            asm volatile("s_wait_dscnt 0x0" ::: "memory");
            async_tile_load(cg + (size_t)(cbase + CT) * EMBD,
                            &c_sh[wave][cur ^ 1][0][0], lane);
            // 16 newer asyncs outstanding -> current buffer's 16 are done
            asm volatile("s_wait_asynccnt 0x10" ::: "memory");
        } else {
            asm volatile("s_wait_asynccnt 0x0" ::: "memory");
        }
        __builtin_amdgcn_wave_barrier();

        // ---- S = Qtile (16xK) * CtileT (Kx16) via fp32 WMMA, K in steps of 4
        // A 16x4 layout: lanes 0-15 hold K=0,1; lanes 16-31 hold K=2,3 (M=lane%16)
        // B 4x16 layout: V0 lanes0-15 K=0 / lanes16-31 K=1; V1: K=2 / K=3 (N=lane%16)
        v8f acc = {};
        #pragma unroll
        for (int k0 = 0; k0 < EMBD; k0 += 4) {
            v2f a, b;
            a.x = q_sh[m16][k0 + 2 * half + 0];
            a.y = q_sh[m16][k0 + 2 * half + 1];
            b.x = c_sh[wave][cur][m16][k0 + half];
            b.y = c_sh[wave][cur][m16][k0 + 2 + half];
            acc = __builtin_amdgcn_wmma_f32_16x16x4_f32(
                false, a, false, b, (short)0, acc, false, false);
        }

        // ---- distances; C/D layout: lane n=lane%16, rows = half*8 + v ----
        const float cnv = cn[cbase + m16];
        #pragma unroll
        for (int v = 0; v < 8; ++v) {
            int r = half * 8 + v;
            float d2 = qn_sh[r] + cnv - 2.0f * acc[v];
            s_sh[wave][r][m16] = sqrtf(fmaxf(d2, 0.0f) + 1e-12f);
        }
        __builtin_amdgcn_wave_barrier();   // intra-wave: DS ops in order per wave

        // ---- 2 lanes per query row scan 8 candidates into register top-8 ---
        {
            int rr   = lane >> 1;
            int cseg = (lane & 1) * 8;
            #pragma unroll
            for (int j = 0; j < 8; ++j) {
                float dv = s_sh[wave][rr][cseg + j];
                int   di = cbase + cseg + j;
                if (dv < topv[TOPN - 1]) {           // rare after warm-up
                    float cv_ = dv; int ci_ = di;
                    #pragma unroll
                    for (int p = 0; p < TOPN; ++p) { // bubble insert (registers)
                        bool sw = cv_ < topv[p];
                        float ov = topv[p]; int oi = topi[p];
                        topv[p] = sw ? cv_ : ov;  topi[p] = sw ? ci_ : oi;
                        cv_     = sw ? ov  : cv_; ci_     = sw ? oi  : ci_;
                    }
                }
            }
        }
        __builtin_amdgcn_wave_barrier();
    }

    // ---- merge per-lane top-8 lists: 64 candidates per query row ----------
    {
        int rr = lane >> 1;
        int base = wave * 16 + (lane & 1) * 8;
        #pragma unroll
        for (int j = 0; j < TOPN; ++j) {
            mv[rr][base + j] = topv[j];
            mi[rr][base + j] = topi[j];
        }
    }
    __syncthreads();

    if (tid < QT) {                                   // one thread per row
        int r = tid;
        for (int s = 0; s < TOPN; ++s) {
            float best = __builtin_inff(); int bj = 0;
            for (int j = 0; j < 64; ++j) {
                float v = mv[r][j];
                if (v < best) { best = v; bj = j; }
            }
            tvv[r][s] = best; tii[r][s] = mi[r][bj];
            mv[r][bj] = 3.4e38f;
        }
    }
    __syncthreads();

    // ---- logits: 128 threads = 16 rows x 8 tops ---------------------------
    {
        int r = tid >> 3, j = tid & 7;
        int idx = tii[r][j];
        const float* cv = cg + (size_t)idx * EMBD;
        float acc = 0.0f;
        #pragma unroll 8
        for (int k = 0; k < EMBD; ++k) acc = fmaf(qp_sh[r][k], cv[k], acc);
        lg[r][j] = acc * 0.08838834764831845f - tvv[r][j];   // /sqrt(128) - dist
    }
    __syncthreads();

    if (tid < QT) {                                   // importance = max softmax
        int r = tid;
        float mx = -3.4e38f;
        #pragma unroll
        for (int j = 0; j < TOPN; ++j) mx = fmaxf(mx, lg[r][j]);
        float ssum = 0.0f;
        #pragma unroll
        for (int j = 0; j < TOPN; ++j) ssum += __expf(lg[r][j] - mx);
        importance[qbase + r] = 1.0f / ssum;          // exp(mx-mx)/sum
    }
}

// --------------------------- final reweighting -----------------------------
__global__ __launch_bounds__(LQ) void reconstruct(const int* __restrict__ qtok,
                                                  const float* __restrict__ importance,
                                                  float* __restrict__ out) {
    __shared__ float red[LQ];
    int b = blockIdx.x, t = threadIdx.x;
    float imp = importance[b * LQ + t];
    red[t] = imp;
    __syncthreads();
    #pragma unroll
    for (int off = 64; off > 0; off >>= 1) {
        if (t < off) red[t] = fmaxf(red[t], red[t + off]);
        __syncthreads();
    }
    float mx = red[0];
    __syncthreads();
    float e = __expf(imp - mx);
    red[t] = e;
    __syncthreads();
    #pragma unroll
    for (int off = 64; off > 0; off >>= 1) {
        if (t < off) red[t] += red[t + off];
        __syncthreads();
    }
    float w = e / red[0];
    out[b * LQ + t] = (float)qtok[b * LQ + t] * w * (float)LQ;
}

// --------------------------- launcher --------------------------------------
extern "C" void kernel_launch(void* const* d_in, const int* in_sizes, int n_in,
                              void* d_out, int out_size, void* d_ws, size_t ws_size,
                              hipStream_t stream) {
    const int*   qtok = (const int*)d_in[0];    // [4,128]
    const int*   ctok = (const int*)d_in[1];    // [65536]
    const float* emb  = (const float*)d_in[2];  // [50257,128]
    const float* W    = (const float*)d_in[3];  // [128,128]
    const float* bias = (const float*)d_in[4];  // [128]
    float*       out  = (float*)d_out;          // [4,128]

    // workspace layout (floats): ~34.2 MB total
    float* ws  = (float*)d_ws;
    float* cg  = ws;                             // 65536*128  (32 MB, L2-resident)
    float* cn  = cg  + (size_t)LC * EMBD;        // 65536
    float* qg  = cn  + LC;                       // 512*128
    float* qn  = qg  + (size_t)NQ * EMBD;        // 512
    float* qp  = qn  + NQ;                       // 512*128
    float* imp = qp  + (size_t)NQ * EMBD;        // 512

    gather_ctx <<<LC / 8, 256, 0, stream>>>(ctok, emb, cg, cn);
    gather_q   <<<NQ, EMBD, 0, stream>>>(qtok, emb, W, bias, qg, qn, qp);
    rag_main   <<<NQ / QT, 128, 0, stream>>>(cg, cn, qg, qn, qp, imp);
    reconstruct<<<BATCH, LQ, 0, stream>>>(qtok, imp, out);
}